// ContrastByClassCalculator_64269890617391
// MI455X (gfx1250) — compile-verified
//
#include <hip/hip_runtime.h>
#include <math.h>

// Problem constants (from reference)
#define DD   128      // feature dim
#define KQ   4096     // queue length
#define CC   100      // classes
#define NN   1024     // samples
#define TILE_N 64     // queue columns per tile
#define NTILES 4      // tiles per block (pipelined)
#define T_INV  14.285714285714286f   // 1/0.07
#define MAXL   T_INV                 // |logit| <= 1/T since logits are cosines
#define EPSN   1e-12f

typedef __attribute__((ext_vector_type(2))) float v2f;
typedef __attribute__((ext_vector_type(8))) float v8f;
typedef __attribute__((ext_vector_type(4))) int   v4i;

// ---- CDNA5 async global->LDS copy (ASYNCcnt) with compile-safe fallback ----
#if __has_builtin(__builtin_amdgcn_global_load_async_to_lds_b128)
#define USE_ASYNC_LDS 1
typedef __attribute__((address_space(1))) v4i* gv4p;   // global int4*
typedef __attribute__((address_space(3))) v4i* lv4p;   // LDS int4* (32-bit)
__device__ __forceinline__ void async_copy_b128(const void* gptr, void* lptr) {
  // int->pointer casts sidestep addrspacecast restrictions; flat LDS address
  // truncated to 32 bits == LDS byte offset (aperture rule)
  __builtin_amdgcn_global_load_async_to_lds_b128(
      (gv4p)(uintptr_t)gptr,
      (lv4p)(uint32_t)(uintptr_t)lptr, 0, 0);
}
#if __has_builtin(__builtin_amdgcn_s_wait_asynccnt)
#define WAIT_ASYNC(n) __builtin_amdgcn_s_wait_asynccnt(n)
#else
#define WAIT_ASYNC(n) asm volatile("s_wait_asynccnt %0" ::"i"(n) : "memory")
#endif
#else
#define USE_ASYNC_LDS 0
#define WAIT_ASYNC(n)
#endif

__device__ __forceinline__ float waveRedSum(float v) {
  #pragma unroll
  for (int m = 16; m >= 1; m >>= 1) v += __shfl_xor(v, m, 32);
  return v;
}
// sums across each 16-lane half of a wave32 (masks < 16 never cross halves)
__device__ __forceinline__ float halfRedSum(float v) {
  #pragma unroll
  for (int m = 8; m >= 1; m >>= 1) v += __shfl_xor(v, m, 32);
  return v;
}

// ---------------------------------------------------------------------------
// Kernel 1: zero accumulators, normalize weight rows -> wn  (grid 100 x 32)
// ---------------------------------------------------------------------------
__global__ void cbc_init(const float* __restrict__ weight, float* __restrict__ wn,
                         float* __restrict__ rowsum, int* __restrict__ cls_count,
                         float* __restrict__ out) {
  int gid = blockIdx.x * 32 + threadIdx.x;
  if (gid < NN) rowsum[gid] = 0.f;
  if (gid < CC) cls_count[gid] = 0;
  if (gid == 0) out[0] = 0.f;

  int c = blockIdx.x;
  int lane = threadIdx.x;
  float w[4]; float ss = 0.f;
  #pragma unroll
  for (int j = 0; j < 4; ++j) { w[j] = weight[c*DD + lane + 32*j]; ss += w[j]*w[j]; }
  ss = waveRedSum(ss);
  float r = 1.f / fmaxf(sqrtf(ss), EPSN);
  #pragma unroll
  for (int j = 0; j < 4; ++j) wn[c*DD + lane + 32*j] = w[j]*r;
}

// ---------------------------------------------------------------------------
// Kernel 2: per-row qa, l_pos, s_n = qa.wn[label]; bucket rows by class.
// One wave per row. grid 256 x 128 (4 waves/block).
// ---------------------------------------------------------------------------
__global__ void cbc_prep(const float* __restrict__ q, const float* __restrict__ k,
                         const int* __restrict__ labels, const float* __restrict__ wn,
                         float* __restrict__ qa, float* __restrict__ s_arr,
                         float* __restrict__ lpos, int* __restrict__ cls_count,
                         int* __restrict__ cls_rows) {
  int wave = threadIdx.x >> 5, lane = threadIdx.x & 31;
  int n = blockIdx.x * 4 + wave;
  if (n >= NN) return;
  int c = labels[n];

  float qv[4], kv[4], wv[4];
  float sq = 0.f, sk = 0.f;
  #pragma unroll
  for (int j = 0; j < 4; ++j) {
    int d = lane + 32*j;
    qv[j] = q[n*DD + d]; kv[j] = k[n*DD + d]; wv[j] = wn[c*DD + d];
    sq += qv[j]*qv[j];   sk += kv[j]*kv[j];
  }
  sq = waveRedSum(sq); sk = waveRedSum(sk);
  float rqn = 1.f / fmaxf(sqrtf(sq), EPSN);
  float rkn = 1.f / fmaxf(sqrtf(sk), EPSN);

  float va[4], vb[4]; float sa = 0.f, sb = 0.f;
  #pragma unroll
  for (int j = 0; j < 4; ++j) {
    va[j] = qv[j]*rqn - wv[j]; sa += va[j]*va[j];
    vb[j] = kv[j]*rkn - wv[j]; sb += vb[j]*vb[j];
  }
  sa = waveRedSum(sa); sb = waveRedSum(sb);
  float ra = 1.f / fmaxf(sqrtf(sa), EPSN);
  float rb = 1.f / fmaxf(sqrtf(sb), EPSN);

  float dot = 0.f, sw = 0.f;
  #pragma unroll
  for (int j = 0; j < 4; ++j) {
    int d = lane + 32*j;
    float aj = va[j]*ra, bj = vb[j]*rb;
    qa[n*DD + d] = aj;
    dot += aj*bj;
    sw  += aj*wv[j];
  }
  dot = waveRedSum(dot); sw = waveRedSum(sw);
  if (lane == 0) {
    lpos[n]  = dot;
    s_arr[n] = sw;
    int pos = atomicAdd(&cls_count[c], 1);
    cls_rows[c*NN + pos] = n;
  }
}

// ---------------------------------------------------------------------------
// Kernel 3: fused queue-streaming GEMM. grid (KQ/(64*4), CC) x 128 threads.
// Block = (class c, 4 x 64-column tiles). Double-buffered LDS tiles filled by
// GLOBAL_LOAD_ASYNC_TO_LDS_B128 while WMMA f32 16x16x4 consumes the other.
// l_neg[n,k] = (P*rq - s_n) * rinv ;  accumulate exp(l_neg/T - MAXL) per row.
// ---------------------------------------------------------------------------
__global__ __launch_bounds__(128)
void cbc_lneg(const float* __restrict__ queue, const float* __restrict__ wn,
              const float* __restrict__ qa, const float* __restrict__ s_arr,
              const int* __restrict__ cls_count, const int* __restrict__ cls_rows,
              float* __restrict__ rowsum) {
  __shared__ float Qsh[2][DD][68];     // double-buffered 128x64 tile; pitch 68
                                       //  -> 16B-aligned rows, +4 bank skew
  __shared__ float Ash[16][132];       // 16 qa rows x 128 depth (padded)
  __shared__ float wnsh[DD];
  __shared__ float rq[TILE_N], rinv[TILE_N];
  __shared__ int   rowsSh[16];

  const int tid  = threadIdx.x;
  const int c    = blockIdx.y;
  const int lane = tid & 31;
  const int wave = tid >> 5;

  if (tid < DD) wnsh[tid] = wn[c*DD + tid];

  const float* qbase = queue + (size_t)c * DD * KQ + blockIdx.x * (TILE_N * NTILES);
  const int r  = tid >> 4;            // 0..7
  const int c4 = (tid & 15) << 2;     // 0,4,...,60

#if USE_ASYNC_LDS
  // issue tile 0 -> buffer 0 (16 x b128 per thread = 32KB per block-tile)
  #pragma unroll 4
  for (int p = 0; p < 16; ++p) {
    int d = p*8 + r;
    async_copy_b128(qbase + (size_t)d * KQ + c4, &Qsh[0][d][c4]);
  }
#endif

  const int count   = cls_count[c];
  const int colBase = wave * 16;
  const int mrow    = lane & 15;
  const int hi      = lane >> 4;              // 0: lanes 0-15, 1: lanes 16-31
  const int col     = colBase + (lane & 15);

  for (int t = 0; t < NTILES; ++t) {
    const int buf = t & 1;
#if USE_ASYNC_LDS
    if (t + 1 < NTILES) {
      // overlap: fill the other buffer while this one is consumed
      const float* nbase = qbase + (t + 1) * TILE_N;
      #pragma unroll 4
      for (int p = 0; p < 16; ++p) {
        int d = p*8 + r;
        async_copy_b128(nbase + (size_t)d * KQ + c4, &Qsh[buf ^ 1][d][c4]);
      }
      WAIT_ASYNC(16);   // tile t's 16 ops retired; tile t+1 still in flight
    } else {
      WAIT_ASYNC(0);
    }
#else
    {
      const float* tbase = qbase + t * TILE_N;
      #pragma unroll 4
      for (int p = 0; p < 16; ++p) {
        int d = p*8 + r;
        if (p < 15) __builtin_prefetch(tbase + (size_t)(d + 8) * KQ + c4, 0, 1);
        const float4 v = *reinterpret_cast<const float4*>(tbase + (size_t)d * KQ + c4);
        Qsh[buf][d][c4+0] = v.x; Qsh[buf][d][c4+1] = v.y;
        Qsh[buf][d][c4+2] = v.z; Qsh[buf][d][c4+3] = v.w;
      }
    }
#endif
    __syncthreads();   // tile t resident in Qsh[buf] for all waves

    // Per-column stats: 1/||col||, rinv = 1/max(sqrt(2-2*wn.qhat), eps)
    if (tid < TILE_N) {
      float ss = 0.f, wd = 0.f;
      #pragma unroll 8
      for (int d = 0; d < DD; ++d) {
        float x = Qsh[buf][d][tid];
        ss += x*x; wd += wnsh[d]*x;
      }
      float rr = 1.f / fmaxf(sqrtf(ss), EPSN);
      float tt = wd * rr;
      float nv = sqrtf(fmaxf(2.f - 2.f*tt, 0.f));
      rq[tid]   = rr;
      rinv[tid] = 1.f / fmaxf(nv, EPSN);
    }
    __syncthreads();

    for (int m0 = 0; m0 < count; m0 += 16) {
      __syncthreads();
      if (tid < 16) rowsSh[tid] = (m0 + tid < count) ? cls_rows[c*NN + m0 + tid] : -1;
      __syncthreads();
      for (int idx = tid; idx < 16*DD; idx += 128) {
        int rr = idx >> 7, d = idx & 127;
        int n = rowsSh[rr];
        Ash[rr][d] = (n >= 0) ? qa[(size_t)n*DD + d] : 0.f;
      }
      __syncthreads();

      v8f acc = {0.f,0.f,0.f,0.f,0.f,0.f,0.f,0.f};
      #pragma unroll 8
      for (int d4 = 0; d4 < DD; d4 += 4) {
        // ISA 16x4 f32 A layout: VGPR0 = K={0|2}, VGPR1 = K={1|3} per half-wave
        v2f a, b;
        a.x = Ash[mrow][d4 + 2*hi];
        a.y = Ash[mrow][d4 + 2*hi + 1];
        b.x = Qsh[buf][d4 + 2*hi][col];
        b.y = Qsh[buf][d4 + 2*hi + 1][col];
        acc = __builtin_amdgcn_wmma_f32_16x16x4_f32(
            false, a, false, b, (short)0, acc, false, false);
      }

      // Epilogue: D layout -> VGPR r holds rows m0+r (lanes 0-15) / m0+r+8
      float rqc = rq[col], ric = rinv[col];
      #pragma unroll
      for (int rr = 0; rr < 8; ++rr) {
        int rloc = rr + 8*hi;
        int n = rowsSh[rloc];              // uniform within each half-wave
        float e = 0.f;
        if (n >= 0) {
          float num = acc[rr]*rqc - s_arr[n];   // qa.qhat - qa.wn
          float lneg = num * ric;
          e = __expf(lneg * T_INV - MAXL);
        }
        e = halfRedSum(e);                 // sum over this wave's 16 columns
        if ((lane & 15) == 0 && n >= 0) unsafeAtomicAdd(&rowsum[n], e);
      }
    }
    __syncthreads();   // all reads of Qsh[buf] done before it is refilled
  }
}

// ---------------------------------------------------------------------------
// Kernel 4: loss = mean(MAXL + log(rowsum + exp(lpos/T - MAXL)) - lpos/T)
// ---------------------------------------------------------------------------
__global__ void cbc_finish(const float* __restrict__ rowsum,
                           const float* __restrict__ lpos, float* __restrict__ out) {
  int n = blockIdx.x * 256 + threadIdx.x;
  float v = 0.f;
  if (n < NN) {
    float lp = lpos[n] * T_INV;
    float S  = rowsum[n] + __expf(lp - MAXL);
    v = (MAXL + __logf(S) - lp) * (1.0f / NN);
  }
  v = waveRedSum(v);
  if ((threadIdx.x & 31) == 0 && v != 0.f) unsafeAtomicAdd(out, v);
}

// ---------------------------------------------------------------------------
extern "C" void kernel_launch(void* const* d_in, const int* in_sizes, int n_in,
                              void* d_out, int out_size, void* d_ws, size_t ws_size,
                              hipStream_t stream) {
  const float* q       = (const float*)d_in[0];
  const float* k       = (const float*)d_in[1];
  const float* weight  = (const float*)d_in[2];
  const int*   labels  = (const int*)d_in[3];
  const float* queue   = (const float*)d_in[4];
  float* out = (float*)d_out;

  // Workspace layout (bytes)
  char* ws = (char*)d_ws;
  float* wn       = (float*)(ws);                     // 100*128
  float* qa       = (float*)(ws + 51200);             // 1024*128
  float* s_arr    = (float*)(ws + 575488);            // 1024
  float* lpos     = (float*)(ws + 579584);            // 1024
  float* rowsum   = (float*)(ws + 583680);            // 1024
  int*   cls_cnt  = (int*)  (ws + 587776);            // 100
  int*   cls_rows = (int*)  (ws + 588176);            // 100*1024

  cbc_init<<<CC, 32, 0, stream>>>(weight, wn, rowsum, cls_cnt, out);
  cbc_prep<<<NN/4, 128, 0, stream>>>(q, k, labels, wn, qa, s_arr, lpos,
                                     cls_cnt, cls_rows);
  dim3 grid(KQ / (TILE_N * NTILES), CC);
  cbc_lneg<<<grid, 128, 0, stream>>>(queue, wn, qa, s_arr, cls_cnt, cls_rows,
                                     rowsum);
  cbc_finish<<<NN/256, 256, 0, stream>>>(rowsum, lpos, out);
}